// GAT_29102698398302
// MI455X (gfx1250) — compile-verified
//
#include <hip/hip_runtime.h>
#include <hip/hip_bf16.h>

#define HEADS 4
#define CH 64
#define HC 256
#define NEG_SLOPE 0.2f

typedef float v2f __attribute__((ext_vector_type(2)));
typedef float v8f __attribute__((ext_vector_type(8)));

// ---- order-preserving float<->uint encoding for atomic max on floats ----
__device__ __forceinline__ unsigned enc_f32(float f) {
    unsigned u = __float_as_uint(f);
    return (u & 0x80000000u) ? ~u : (u | 0x80000000u);
}
__device__ __forceinline__ float dec_f32(unsigned u) {
    return (u & 0x80000000u) ? __uint_as_float(u ^ 0x80000000u)
                             : __uint_as_float(~u);
}
#define ENC_NEG_INF 0x007FFFFFu   // enc(-inf)

__device__ __forceinline__ void edge_sd(const int* ei, int E, int e, int& s, int& d) {
    if (e < E) { s = ei[e]; d = ei[E + e]; }
    else       { s = e - E; d = s; }
}

// ------------------------- utility kernels -------------------------
__global__ void k_zero(float* p, long n) {
    long i = (long)blockIdx.x * blockDim.x + threadIdx.x;
    long st = (long)gridDim.x * blockDim.x;
    for (; i < n; i += st) p[i] = 0.0f;
}

__global__ void k_init_softmax(unsigned* amax, float* denom, long n) {
    long i = (long)blockIdx.x * blockDim.x + threadIdx.x;
    long st = (long)gridDim.x * blockDim.x;
    for (; i < n; i += st) { amax[i] = ENC_NEG_INF; denom[i] = 0.0f; }
}

__global__ void k_sum(const float* p, int n, float* out) {
    long i = (long)blockIdx.x * blockDim.x + threadIdx.x;
    long st = (long)gridDim.x * blockDim.x;
    float s = 0.0f;
    for (; i < n; i += st) s += p[i];
    atomicAdd(out, s);
}

// coef[h] = sum_c We[h*CH+c] * att_e[h*CH+c]
__global__ void k_coef(const float* We, const float* ae, float* coef) {
    int hh = threadIdx.x;
    if (hh < HEADS) {
        float s = 0.0f;
        for (int c = 0; c < CH; ++c) s += We[hh * CH + c] * ae[hh * CH + c];
        coef[hh] = s;
    }
}

// --------------------- fp32 WMMA GEMM: C[M,Nc] = A[M,K] @ B[K,Nc] ---------------------
// One wave32 computes a 16x64 output strip (4 N-tiles): a single A fragment per
// K-step feeds 4 back-to-back V_WMMA_F32_16X16X4_F32 ops (A-reuse, dense WMMA
// issue). All indexing is int32 pointer-bumping to avoid 64-bit address VALU.
__global__ void k_gemm_wmma_f32(const float* __restrict__ A, const float* __restrict__ B,
                                float* __restrict__ Cout, int M, int K, int Nc) {
    int wave = (int)((blockIdx.x * blockDim.x + threadIdx.x) >> 5);
    int lane = threadIdx.x & 31;
    int ngroups = Nc >> 6;                     // 64-wide column groups
    int mt = wave / ngroups;
    int ng = wave - mt * ngroups;
    if (mt * 16 >= M) return;                  // wave-uniform: EXEC stays all-ones

    int l16  = lane & 15;
    int half = lane >> 4;                      // 0 or 1
    int row  = mt * 16 + l16;
    if (row >= M) row = M - 1;                 // clamp (M%16==0 in practice)

    // A fragment: lane l -> row l%16, K = k0 + 2*(l/16) + v
    const float* Ap = A + row * K + 2 * half;
    // B fragment: lane l -> col base + l%16, K = k0 + 2*(l/16) + v
    const float* Bp = B + (2 * half) * Nc + ng * 64 + l16;

    v8f acc0 = {}, acc1 = {}, acc2 = {}, acc3 = {};
    for (int k0 = 0; k0 < K; k0 += 4) {
        v2f a;
        a.x = Ap[0];
        a.y = Ap[1];
        v2f b0, b1, b2, b3;
        b0.x = Bp[0];       b0.y = Bp[Nc];
        b1.x = Bp[16];      b1.y = Bp[Nc + 16];
        b2.x = Bp[32];      b2.y = Bp[Nc + 32];
        b3.x = Bp[48];      b3.y = Bp[Nc + 48];
        acc0 = __builtin_amdgcn_wmma_f32_16x16x4_f32(false, a, false, b0, (short)0, acc0, false, false);
        acc1 = __builtin_amdgcn_wmma_f32_16x16x4_f32(false, a, false, b1, (short)0, acc1, false, false);
        acc2 = __builtin_amdgcn_wmma_f32_16x16x4_f32(false, a, false, b2, (short)0, acc2, false, false);
        acc3 = __builtin_amdgcn_wmma_f32_16x16x4_f32(false, a, false, b3, (short)0, acc3, false, false);
        Ap += 4;
        Bp += 4 * Nc;
    }

    // C/D layout: lane l, vgpr v -> n = l%16, m = v + 8*(l/16)
    int cmBase = mt * 16 + 8 * half;
    float* Cp = Cout + cmBase * Nc + ng * 64 + l16;
#pragma unroll
    for (int v = 0; v < 8; ++v) {
        int m = cmBase + v;
        if (m < M) {
            Cp[0]  = acc0[v];
            Cp[16] = acc1[v];
            Cp[32] = acc2[v];
            Cp[48] = acc3[v];
        }
        Cp += Nc;
    }
}

// ------------------- per-node attention dot products -------------------
__global__ void k_node_attn(const float* __restrict__ h,
                            const float* __restrict__ att_s, const float* __restrict__ att_d,
                            float* __restrict__ a_src, float* __restrict__ a_dst, int N) {
    int idx = blockIdx.x * blockDim.x + threadIdx.x;
    if (idx >= N * HEADS) return;
    int n = idx >> 2, hh = idx & 3;
    const float* hv = h + (long)n * HC + hh * CH;
    float s = 0.0f, d = 0.0f;
#pragma unroll 8
    for (int c = 0; c < CH; ++c) {
        float v = hv[c];
        s += v * att_s[hh * CH + c];
        d += v * att_d[hh * CH + c];
    }
    a_src[idx] = s;
    a_dst[idx] = d;
}

// ------------------- edge pass 1: alpha + segment max -------------------
__global__ void k_edge_alpha(const int* __restrict__ ei, int E, int Et,
                             const float* __restrict__ ea, const float* __restrict__ easum,
                             float invE, const float* __restrict__ coef,
                             const float* __restrict__ a_src, const float* __restrict__ a_dst,
                             float* __restrict__ alpha, unsigned* __restrict__ amax) {
    int e = blockIdx.x * blockDim.x + threadIdx.x;
    if (e >= Et) return;
    int s, d;
    edge_sd(ei, E, e, s, d);
    float aval = (e < E) ? ea[e] : easum[0] * invE;
#pragma unroll
    for (int hh = 0; hh < HEADS; ++hh) {
        float al = a_src[s * HEADS + hh] + a_dst[d * HEADS + hh] + aval * coef[hh];
        al = (al > 0.0f) ? al : NEG_SLOPE * al;
        alpha[(long)e * HEADS + hh] = al;
        atomicMax(&amax[d * HEADS + hh], enc_f32(al));
    }
}

// ------------------- edge pass 2: exp + segment sum -------------------
__global__ void k_edge_exp(const int* __restrict__ ei, int E, int Et,
                           float* __restrict__ alpha, const unsigned* __restrict__ amax,
                           float* __restrict__ denom) {
    int e = blockIdx.x * blockDim.x + threadIdx.x;
    if (e >= Et) return;
    int s, d;
    edge_sd(ei, E, e, s, d);
#pragma unroll
    for (int hh = 0; hh < HEADS; ++hh) {
        float m = dec_f32(amax[d * HEADS + hh]);
        float ex = __expf(alpha[(long)e * HEADS + hh] - m);
        alpha[(long)e * HEADS + hh] = ex;       // overwrite with exp value
        atomicAdd(&denom[d * HEADS + hh], ex);
    }
}

// ------------------- edge pass 3: weighted scatter-add -------------------
// One 256-thread block per edge; thread = output channel. out[dst] (51 MB) and
// h (51 MB) both fit in the 192 MB L2, so the atomics stay on-chip.
__global__ void k_edge_scatter(const int* __restrict__ ei, int E, int Et,
                               const float* __restrict__ ex, const float* __restrict__ denom,
                               const float* __restrict__ h, float* __restrict__ out) {
    int e = blockIdx.x;
    if (e >= Et) return;
    int c = threadIdx.x;
    int hh = c >> 6;
    int s, d;
    edge_sd(ei, E, e, s, d);
    float a = ex[(long)e * HEADS + hh] / (denom[d * HEADS + hh] + 1e-16f);
    atomicAdd(&out[(long)d * HC + c], a * h[(long)s * HC + c]);
}

__global__ void k_bias(float* __restrict__ out, const float* __restrict__ b, long n) {
    long i = (long)blockIdx.x * blockDim.x + threadIdx.x;
    long st = (long)gridDim.x * blockDim.x;
    for (; i < n; i += st) out[i] += b[i & (HC - 1)];
}

// ----------------------------------------------------------------------
extern "C" void kernel_launch(void* const* d_in, const int* in_sizes, int n_in,
                              void* d_out, int out_size, void* d_ws, size_t ws_size,
                              hipStream_t stream) {
    const float* x   = (const float*)d_in[0];
    const int*   ei  = (const int*)d_in[1];
    // d_in[2] = batch (unused: single graph)
    const float* ea  = (const float*)d_in[3];
    const float* W1  = (const float*)d_in[4];
    const float* We1 = (const float*)d_in[5];
    const float* as1 = (const float*)d_in[6];
    const float* ad1 = (const float*)d_in[7];
    const float* ae1 = (const float*)d_in[8];
    const float* b1  = (const float*)d_in[9];
    const float* W2  = (const float*)d_in[10];
    const float* We2 = (const float*)d_in[11];
    const float* as2 = (const float*)d_in[12];
    const float* ad2 = (const float*)d_in[13];
    const float* ae2 = (const float*)d_in[14];
    const float* b2  = (const float*)d_in[15];

    const int N  = in_sizes[2];        // batch has N entries
    const int E  = in_sizes[3];        // edge_attr has E entries (1 per edge)
    const int IN = in_sizes[0] / N;    // 128
    const int Et = E + N;              // with self-loops

    // ---- workspace carve-up (floats) ----
    float* ws = (float*)d_ws;
    size_t off = 0;
    float*    h     = ws + off; off += (size_t)N * HC;     // GEMM output (per layer, reused)
    float*    hout1 = ws + off; off += (size_t)N * HC;     // layer-1 aggregated output
    float*    asrc  = ws + off; off += (size_t)N * HEADS;
    float*    adst  = ws + off; off += (size_t)N * HEADS;
    unsigned* amax  = (unsigned*)(ws + off); off += (size_t)N * HEADS;
    float*    denom = ws + off; off += (size_t)N * HEADS;
    float*    alpha = ws + off; off += (size_t)Et * HEADS; // alpha -> exp, in place
    float*    coef1 = ws + off; off += 4;
    float*    coef2 = ws + off; off += 4;
    float*    easum = ws + off; off += 1;

    float* dout = (float*)d_out;

    const int T = 256;
    const long nNodeF = (long)N * HC;
    const long nSoft  = (long)N * HEADS;
    int gZero  = 2048;
    int gEdge  = (Et + T - 1) / T;
    int gAttn  = (N * HEADS + T - 1) / T;
    int gSum   = 1024;

    auto gemm_grid = [&](int M, int Nc) {
        long waves = (long)((M + 15) / 16) * (Nc / 64);   // 16x64 strip per wave
        return (int)((waves * 32 + T - 1) / T);
    };

    // ---- init accumulators (fresh every call: graph-replay safe) ----
    k_zero<<<gZero, T, 0, stream>>>(hout1, nNodeF);
    k_zero<<<gZero, T, 0, stream>>>(dout, nNodeF);
    k_zero<<<1, 64, 0, stream>>>(easum, 1);
    k_sum<<<gSum, T, 0, stream>>>(ea, E, easum);
    k_coef<<<1, 64, 0, stream>>>(We1, ae1, coef1);
    k_coef<<<1, 64, 0, stream>>>(We2, ae2, coef2);

    // =================== layer 1 ===================
    k_gemm_wmma_f32<<<gemm_grid(N, HC), T, 0, stream>>>(x, W1, h, N, IN, HC);
    k_node_attn<<<gAttn, T, 0, stream>>>(h, as1, ad1, asrc, adst, N);
    k_init_softmax<<<gZero, T, 0, stream>>>(amax, denom, nSoft);
    k_edge_alpha<<<gEdge, T, 0, stream>>>(ei, E, Et, ea, easum, 1.0f / (float)E,
                                          coef1, asrc, adst, alpha, amax);
    k_edge_exp<<<gEdge, T, 0, stream>>>(ei, E, Et, alpha, amax, denom);
    k_edge_scatter<<<Et, HC, 0, stream>>>(ei, E, Et, alpha, denom, h, hout1);
    k_bias<<<gZero, T, 0, stream>>>(hout1, b1, nNodeF);

    // =================== layer 2 ===================
    k_gemm_wmma_f32<<<gemm_grid(N, HC), T, 0, stream>>>(hout1, W2, h, N, HC, HC);
    k_node_attn<<<gAttn, T, 0, stream>>>(h, as2, ad2, asrc, adst, N);
    k_init_softmax<<<gZero, T, 0, stream>>>(amax, denom, nSoft);
    k_edge_alpha<<<gEdge, T, 0, stream>>>(ei, E, Et, ea, easum, 1.0f / (float)E,
                                          coef2, asrc, adst, alpha, amax);
    k_edge_exp<<<gEdge, T, 0, stream>>>(ei, E, Et, alpha, amax, denom);
    k_edge_scatter<<<Et, HC, 0, stream>>>(ei, E, Et, alpha, denom, h, dout);
    k_bias<<<gZero, T, 0, stream>>>(dout, b2, nNodeF);
}